// ChaoticFeatureExtractor_71683004170600
// MI455X (gfx1250) — compile-verified
//
#include <hip/hip_runtime.h>
#include <math.h>

typedef float v2f __attribute__((ext_vector_type(2)));
typedef float v8f __attribute__((ext_vector_type(8)));

#define SLEN 2048
#define NB   16
#define DIN  256
#define OUTF 192

// workspace layout (float offsets)
#define OFF_E    0                       // (B*S, 8): [0:5]=t, [5:8]=r
#define OFF_SQ   (OFF_E + NB*SLEN*8)     // (B*S)
#define OFF_MAX  (OFF_SQ + NB*SLEN)      // 16: per-batch max d2 (as float bits, atomicMax-able since >=0)
#define OFF_SUMR (OFF_MAX + 16)          // 16
#define OFF_DET  (OFF_SUMR + 16)         // 16
#define OFF_VERT (OFF_DET + 16)          // 16
#define OFF_FV   (OFF_VERT + 16)         // 32: per-batch (mean, std) of log-distances

__global__ void k_init(float* __restrict__ ws) {
    int t = threadIdx.x;
    if (t < 64) ws[OFF_MAX + t] = 0.0f;
}

// ---------------------------------------------------------------------------
// Kernel 1: fused first+second layer embeddings via V_WMMA_F32_16X16X4_F32.
// Combined hidden = 16 cols (10 mle + 6 rqa).  One wave per 16-row block.
// ---------------------------------------------------------------------------
__global__ __launch_bounds__(256) void k_embed(
    const float* __restrict__ x,
    const float* __restrict__ mW1, const float* __restrict__ mb1,
    const float* __restrict__ mW2, const float* __restrict__ mb2,
    const float* __restrict__ rW1, const float* __restrict__ rb1,
    const float* __restrict__ rW2, const float* __restrict__ rb2,
    float* __restrict__ ws)
{
    __shared__ float w1s[256 * 16];   // combined W1 (K=256, N=16)
    __shared__ float w2s[16 * 8];     // combined W2' (16 -> 8), zero-padded blocks
    __shared__ float b1s[16];
    __shared__ float b2s[8];
    __shared__ float hs[8][16][17];   // per-wave transpose buffer (padded)

    int tid = threadIdx.x;
    for (int idx = tid; idx < 4096; idx += 256) {
        int k = idx >> 4, n = idx & 15;
        w1s[idx] = (n < 10) ? mW1[k * 10 + n] : rW1[k * 6 + (n - 10)];
    }
    for (int idx = tid; idx < 128; idx += 256) {
        int k = idx >> 3, n = idx & 7;
        float v = 0.0f;
        if (k < 10 && n < 5)        v = mW2[k * 5 + n];
        else if (k >= 10 && n >= 5) v = rW2[(k - 10) * 3 + (n - 5)];
        w2s[idx] = v;
    }
    if (tid < 16) b1s[tid] = (tid < 10) ? mb1[tid] : rb1[tid - 10];
    if (tid < 8)  b2s[tid] = (tid < 5)  ? mb2[tid] : rb2[tid - 5];
    __syncthreads();

    int wid  = tid >> 5;
    int lane = tid & 31;
    int rb   = blockIdx.x * 8 + wid;     // 2048 row-blocks total
    int g0   = rb * 16;
    int m    = lane & 15;                // A: row;  B: col
    int kh   = (lane >> 4) * 2;          // K half select (K=kh, kh+1)

    const float* xrow = x + (size_t)(g0 + m) * DIN + kh;
    v8f c = {0.f, 0.f, 0.f, 0.f, 0.f, 0.f, 0.f, 0.f};
    #pragma unroll 4
    for (int k = 0; k < 64; ++k) {
        float2 av = *(const float2*)(xrow + 4 * k);
        v2f a;  a[0] = av.x;  a[1] = av.y;
        v2f bb;
        bb[0] = w1s[(4 * k + kh) * 16 + m];
        bb[1] = w1s[(4 * k + kh + 1) * 16 + m];
        c = __builtin_amdgcn_wmma_f32_16x16x4_f32(false, a, false, bb,
                                                  (short)0, c, false, false);
    }
    // bias + ReLU, transpose via LDS (C layout: lane=col m, rows v+8*(lane>>4))
    float bias = b1s[m];
    int rbase = 8 * (lane >> 4);
    #pragma unroll
    for (int v = 0; v < 8; ++v)
        hs[wid][rbase + v][m] = fmaxf(c[v] + bias, 0.0f);
    __syncthreads();

    // second (linear) layer: lane handles row m, outputs n0..n0+3
    int n0 = (lane >> 4) * 4;
    float a0 = b2s[n0 + 0], a1 = b2s[n0 + 1], a2 = b2s[n0 + 2], a3 = b2s[n0 + 3];
    #pragma unroll
    for (int k = 0; k < 16; ++k) {
        float hv = hs[wid][m][k];
        a0 += hv * w2s[k * 8 + n0 + 0];
        a1 += hv * w2s[k * 8 + n0 + 1];
        a2 += hv * w2s[k * 8 + n0 + 2];
        a3 += hv * w2s[k * 8 + n0 + 3];
    }
    float* e = ws + OFF_E;
    *(float4*)(e + (size_t)(g0 + m) * 8 + n0) = make_float4(a0, a1, a2, a3);
    if (n0 == 4)  // this half holds e[4..7] = (t4, r0, r1, r2)
        ws[OFF_SQ + g0 + m] = a1 * a1 + a2 * a2 + a3 * a3;
}

// ---------------------------------------------------------------------------
// Kernel 2: per-batch MLE stats: mean / unbiased std of log(||t[i+2]-t[i]||+eps)
// ---------------------------------------------------------------------------
__global__ __launch_bounds__(256) void k_mle(float* __restrict__ ws)
{
    __shared__ float s1[256], s2[256];
    int b = blockIdx.x, tid = threadIdx.x;
    const float* e = ws + OFF_E;
    float sum = 0.f, sumsq = 0.f;
    for (int i = tid; i < SLEN - 2; i += 256) {
        const float* t0 = e + (size_t)(b * SLEN + i) * 8;
        const float* t2 = e + (size_t)(b * SLEN + i + 2) * 8;
        float d2 = 0.f;
        #pragma unroll
        for (int c = 0; c < 5; ++c) { float d = t2[c] - t0[c]; d2 += d * d; }
        float ld = logf(sqrtf(d2) + 1e-6f);
        sum += ld;  sumsq += ld * ld;
    }
    s1[tid] = sum;  s2[tid] = sumsq;  __syncthreads();
    for (int off = 128; off > 0; off >>= 1) {
        if (tid < off) { s1[tid] += s1[tid + off]; s2[tid] += s2[tid + off]; }
        __syncthreads();
    }
    if (tid == 0) {
        float n = (float)(SLEN - 2);
        float mean = s1[0] / n;
        float var  = fmaxf((s2[0] - n * mean * mean) / (n - 1.0f), 0.0f);
        ws[OFF_FV + b * 2 + 0] = mean;
        ws[OFF_FV + b * 2 + 1] = sqrtf(var);
    }
}

// ---------------------------------------------------------------------------
// Kernel 3: per-batch max of clamped pairwise d2 (WMMA dot products, K=3 pad 4)
// one wave per 16-column strip; 128 row tiles each
// ---------------------------------------------------------------------------
__global__ __launch_bounds__(128) void k_maxd2(float* __restrict__ ws)
{
    const float* e  = ws + OFF_E;
    const float* sq = ws + OFF_SQ;
    int tid = threadIdx.x, lane = tid & 31, wid = tid >> 5;
    int strip = blockIdx.x * 4 + wid;       // 0..2047
    int b = strip >> 7, jt = strip & 127;
    int col = lane & 15, kh = (lane >> 4) * 2, h = lane >> 4;
    int bq = b * SLEN;
    int j = bq + jt * 16 + col;
    v2f bb;
    bb[0] = e[(size_t)j * 8 + 5 + kh];
    bb[1] = (kh == 0) ? e[(size_t)j * 8 + 6] : 0.0f;
    float sqj = sq[j];
    float mx = 0.0f;
    for (int it = 0; it < 128; ++it) {
        int i0 = it * 16;
        int i = bq + i0 + col;
        v2f a;
        a[0] = e[(size_t)i * 8 + 5 + kh];
        a[1] = (kh == 0) ? e[(size_t)i * 8 + 6] : 0.0f;
        v8f c = {0.f, 0.f, 0.f, 0.f, 0.f, 0.f, 0.f, 0.f};
        c = __builtin_amdgcn_wmma_f32_16x16x4_f32(false, a, false, bb,
                                                  (short)0, c, false, false);
        const float4* sp = (const float4*)(sq + bq + i0 + 8 * h);
        float4 sA = sp[0], sB = sp[1];
        float sqa[8] = {sA.x, sA.y, sA.z, sA.w, sB.x, sB.y, sB.z, sB.w};
        #pragma unroll
        for (int v = 0; v < 8; ++v)
            mx = fmaxf(mx, sqa[v] + sqj - 2.0f * c[v]);
    }
    for (int off = 16; off >= 1; off >>= 1)
        mx = fmaxf(mx, __shfl_xor(mx, off, 32));
    if (lane == 0)
        atomicMax((unsigned int*)(ws + OFF_MAX + b), __float_as_uint(fmaxf(mx, 0.0f)));
}

// ---------------------------------------------------------------------------
// Kernel 4: RQA statistics (sumR, det numerator, vert numerator) without
// materializing R.  Column-run [0,1,1] pattern counted with bitmask popcounts.
// ---------------------------------------------------------------------------
__global__ __launch_bounds__(128) void k_rqa(const float* __restrict__ thrp,
                                             float* __restrict__ ws)
{
    const float* e  = ws + OFF_E;
    const float* sq = ws + OFF_SQ;
    int tid = threadIdx.x, lane = tid & 31, wid = tid >> 5;
    int strip = blockIdx.x * 4 + wid;
    int b = strip >> 7, jt = strip & 127;
    int col = lane & 15, kh = (lane >> 4) * 2, h = lane >> 4;
    int bq = b * SLEN;
    int jg = jt * 16 + col;                 // column index within batch
    int j = bq + jg;
    v2f bb;
    bb[0] = e[(size_t)j * 8 + 5 + kh];
    bb[1] = (kh == 0) ? e[(size_t)j * 8 + 6] : 0.0f;
    float sqj = sq[j];
    float sig = 1.0f / (1.0f + expf(-thrp[0]));
    float thr2 = sig * sig * ws[OFF_MAX + b];   // dist<thr  <=>  clamp(d2)<thr2
    unsigned carry = 0u;
    int sumR = 0, det = 0, vert = 0;
    for (int it = 0; it < 128; ++it) {
        int i0 = it * 16;
        int i = bq + i0 + col;
        v2f a;
        a[0] = e[(size_t)i * 8 + 5 + kh];
        a[1] = (kh == 0) ? e[(size_t)i * 8 + 6] : 0.0f;
        v8f c = {0.f, 0.f, 0.f, 0.f, 0.f, 0.f, 0.f, 0.f};
        c = __builtin_amdgcn_wmma_f32_16x16x4_f32(false, a, false, bb,
                                                  (short)0, c, false, false);
        const float4* sp = (const float4*)(sq + bq + i0 + 8 * h);
        float4 sA = sp[0], sB = sp[1];
        float sqa[8] = {sA.x, sA.y, sA.z, sA.w, sB.x, sB.y, sB.z, sB.w};
        unsigned mask = 0u;
        int ibase = i0 + 8 * h;
        #pragma unroll
        for (int v = 0; v < 8; ++v) {
            float d2 = fmaxf(sqa[v] + sqj - 2.0f * c[v], 0.0f);
            unsigned bit = (d2 < thr2) ? 1u : 0u;
            mask |= bit << v;
            int dji = jg - (ibase + v);
            det += (int)(bit & ((dji >= 1 && dji <= 9) ? 1u : 0u));
        }
        sumR += __popc((int)mask);
        // column-run pattern [prev=0, cur=1, next=1], anchored at pair-end so
        // only past history is needed: z = {prev2, prev1, rows 0..7}
        unsigned pair = (unsigned)__shfl_xor((int)mask, 16, 32);
        unsigned hist = (h == 0) ? carry : pair;  // h=1 sees current h=0 mask
        unsigned z = (mask << 2) | ((hist >> 6) & 3u);
        unsigned q = z & (z << 1) & ~(z << 2);
        vert += __popc((int)(q & 0x3FCu));
        carry = pair;                       // next tile's h=0 needs this h=1 mask
    }
    for (int off = 16; off >= 1; off >>= 1) {
        sumR += __shfl_xor(sumR, off, 32);
        det  += __shfl_xor(det,  off, 32);
        vert += __shfl_xor(vert, off, 32);
    }
    if (lane == 0) {
        atomicAdd(ws + OFF_SUMR + b, (float)sumR);
        atomicAdd(ws + OFF_DET  + b, (float)det);
        atomicAdd(ws + OFF_VERT + b, (float)vert);
    }
}

// ---------------------------------------------------------------------------
// Kernel 5: metrics -> branch heads -> fusion GEMM -> ReLU -> BatchNorm(train)
// one workgroup; thread o owns output feature o across the batch.
// ---------------------------------------------------------------------------
__global__ __launch_bounds__(256) void k_final(
    const float* __restrict__ mWe, const float* __restrict__ mbe,
    const float* __restrict__ rWr, const float* __restrict__ rbr,
    const float* __restrict__ fW,  const float* __restrict__ fb,
    const float* __restrict__ gamma, const float* __restrict__ beta,
    const float* __restrict__ ws, float* __restrict__ out)
{
    __shared__ float cbuf[NB][OUTF + 1];
    __shared__ float met[NB][4];
    int tid = threadIdx.x;
    if (tid < NB) {
        int b = tid;
        float sumR = ws[OFF_SUMR + b];
        float den = sumR + 1e-6f;
        met[b][0] = sumR / (float)((float)SLEN * (float)SLEN);
        met[b][1] = ws[OFF_DET + b] / den;
        met[b][2] = ws[OFF_VERT + b] / den;
        met[b][3] = -sumR * logf(1.0f + 1e-6f);   // entr: R in {0,1}
    }
    __syncthreads();
    for (int idx = tid; idx < NB * OUTF; idx += 256) {
        int b = idx / OUTF, o = idx % OUTF;
        float v;
        if (o < 64) {
            float f0 = ws[OFF_FV + b * 2], f1 = ws[OFF_FV + b * 2 + 1];
            v = tanhf(f0 * mWe[o] + f1 * mWe[64 + o] + mbe[o]);
        } else {
            int o2 = o - 64;
            float a = rbr[o2];
            #pragma unroll
            for (int k = 0; k < 4; ++k) a += met[b][k] * rWr[k * 128 + o2];
            v = fmaxf(a, 0.0f);
        }
        cbuf[b][o] = v;
    }
    __syncthreads();
    if (tid < OUTF) {
        int o = tid;
        float hreg[NB];
        float mu = 0.f;
        for (int b = 0; b < NB; ++b) {
            float a = fb[o];
            for (int k = 0; k < OUTF; ++k) a += cbuf[b][k] * fW[k * OUTF + o];
            a = fmaxf(a, 0.0f);
            hreg[b] = a;  mu += a;
        }
        mu *= (1.0f / (float)NB);
        float var = 0.f;
        for (int b = 0; b < NB; ++b) { float d = hreg[b] - mu; var += d * d; }
        var *= (1.0f / (float)NB);
        float inv = rsqrtf(var + 1e-5f);
        float g = gamma[o], be = beta[o];
        for (int b = 0; b < NB; ++b)
            out[b * OUTF + o] = g * (hreg[b] - mu) * inv + be;
    }
}

extern "C" void kernel_launch(void* const* d_in, const int* in_sizes, int n_in,
                              void* d_out, int out_size, void* d_ws, size_t ws_size,
                              hipStream_t stream)
{
    const float* x    = (const float*)d_in[0];
    const float* mW1  = (const float*)d_in[1];
    const float* mb1  = (const float*)d_in[2];
    const float* mW2  = (const float*)d_in[3];
    const float* mb2  = (const float*)d_in[4];
    const float* mWe  = (const float*)d_in[5];
    const float* mbe  = (const float*)d_in[6];
    const float* rW1  = (const float*)d_in[7];
    const float* rb1  = (const float*)d_in[8];
    const float* rW2  = (const float*)d_in[9];
    const float* rb2  = (const float*)d_in[10];
    const float* thr  = (const float*)d_in[11];
    const float* rWr  = (const float*)d_in[12];
    const float* rbr  = (const float*)d_in[13];
    const float* fW   = (const float*)d_in[14];
    const float* fb   = (const float*)d_in[15];
    const float* gam  = (const float*)d_in[16];
    const float* bet  = (const float*)d_in[17];
    float* ws  = (float*)d_ws;
    float* out = (float*)d_out;

    k_init <<<dim3(1),   dim3(64),  0, stream>>>(ws);
    k_embed<<<dim3(256), dim3(256), 0, stream>>>(x, mW1, mb1, mW2, mb2,
                                                 rW1, rb1, rW2, rb2, ws);
    k_mle  <<<dim3(16),  dim3(256), 0, stream>>>(ws);
    k_maxd2<<<dim3(512), dim3(128), 0, stream>>>(ws);
    k_rqa  <<<dim3(512), dim3(128), 0, stream>>>(thr, ws);
    k_final<<<dim3(1),   dim3(256), 0, stream>>>(mWe, mbe, rWr, rbr, fW, fb,
                                                 gam, bet, ws, out);
}